// TopKActivation_54924041781827
// MI455X (gfx1250) — compile-verified
//
#include <hip/hip_runtime.h>

// TopK row activation for MI455X (gfx1250):
//  - one workgroup per row, row staged once into LDS via the Tensor Data Mover
//  - exact 4x8-bit radix select (histogram in LDS via ds_add atomics)
//  - b128 LDS reads + b128 non-temporal global stores
//  - single HBM read + single HBM write => bandwidth-bound floor ~22us @ 23.3 TB/s

static constexpr int DIM   = 16384;   // row length
static constexpr int KEEP  = 512;     // K
static constexpr int BLOCK = 512;     // 16 wave32s
static constexpr int VPT   = (DIM / 4) / BLOCK;  // 8 float4 per thread

typedef __attribute__((ext_vector_type(4))) float        vf4;   // native vector (NT-store capable)
typedef __attribute__((ext_vector_type(4))) unsigned int tdm_v4u;
typedef __attribute__((ext_vector_type(8))) int          tdm_v8i;
typedef __attribute__((ext_vector_type(4))) int          tdm_v4i;

// Order-preserving map: larger float <=> larger unsigned key (NaN-free inputs).
__device__ __forceinline__ unsigned int f2key(float f) {
  unsigned int b = __float_as_uint(f);
  return (b & 0x80000000u) ? ~b : (b | 0x80000000u);
}

__global__ __launch_bounds__(BLOCK)
void topk_mask_kernel(const float* __restrict__ x, float* __restrict__ out) {
  __shared__ __align__(16) float row_f[DIM];   // 64 KB row tile
  __shared__ unsigned int hist[256];
  __shared__ unsigned int sfx[257];            // suffix counts
  __shared__ unsigned int bc[4];               // [0]=prefix [1]=remaining [2]=eq_count [3]=cut_idx

  const unsigned int tid = threadIdx.x;
  const unsigned int row = blockIdx.x;
  const float* __restrict__ grow = x + (size_t)row * DIM;

  // ---------------- Stage 1: stream row into LDS via TDM ----------------
#if defined(__gfx1250__) && __has_builtin(__builtin_amdgcn_tensor_load_to_lds)
  if (tid < 32u) {  // one wave issues the TDM descriptor
    unsigned long long ga = (unsigned long long)(const void*)grow;
    // Low 32 bits of a flat LDS pointer are the LDS byte offset.
    unsigned int lds_addr = (unsigned int)(unsigned long long)(void*)row_f;

    tdm_v4u g0;
    g0[0] = 1u;                                    // count=1 valid descriptor, no gather
    g0[1] = lds_addr;                              // lds_addr[31:0]
    g0[2] = (unsigned int)ga;                      // global_addr[31:0]
    g0[3] = (unsigned int)((ga >> 32) & 0x01FFFFFFull) | (2u << 30); // addr[56:32] | type=2

    tdm_v8i g1;
    g1[0] = (int)(2u << 16);                       // workgroup_mask=0, data_size=2 (4 bytes)
    g1[1] = (int)(((unsigned)DIM & 0xFFFFu) << 16);// tensor_dim0[15:0] << 16
    g1[2] = (int)(((unsigned)DIM >> 16) | (1u << 16)); // tensor_dim0[31:16] | tensor_dim1[15:0]=1
    g1[3] = (int)(((unsigned)DIM & 0xFFFFu) << 16);// tensor_dim1[31:16]=0 | tile_dim0=DIM
    g1[4] = 1;                                     // tile_dim1=1, tile_dim2=0
    g1[5] = DIM;                                   // tensor_dim0_stride[31:0]
    g1[6] = (int)(((unsigned)DIM & 0xFFFFu) << 16);// stride0[47:32]=0 | tensor_dim1_stride[15:0]
    g1[7] = 0;                                     // tensor_dim1_stride[47:16]=0

    tdm_v4i g2 = {0, 0, 0, 0};                     // <=2D tensor: groups 2/3 unused
    tdm_v4i g3 = {0, 0, 0, 0};
    tdm_v8i g4 = {0, 0, 0, 0, 0, 0, 0, 0};
    __builtin_amdgcn_tensor_load_to_lds(g0, g1, g2, g3, g4, 0);
  }
  #if __has_builtin(__builtin_amdgcn_s_wait_tensorcnt)
  __builtin_amdgcn_s_wait_tensorcnt(0);
  #else
  asm volatile("s_wait_tensorcnt 0x0" ::: "memory");
  #endif
#else
  {
    const vf4* g4 = reinterpret_cast<const vf4*>(grow);
    vf4* l4w = reinterpret_cast<vf4*>(row_f);
    #pragma unroll
    for (int j = 0; j < VPT; ++j)
      l4w[tid + j * BLOCK] = g4[tid + j * BLOCK];
  }
#endif
  __syncthreads();

  const vf4* __restrict__ l4 = reinterpret_cast<const vf4*>(row_f);

  // ---------------- Stage 2: exact radix select of K-th largest key ----------------
  unsigned int prefix = 0;      // known high bits of the K-th key
  unsigned int rem    = KEEP;   // rank remaining within current prefix class

  #pragma unroll
  for (int p = 0; p < 4; ++p) {
    const int shift = 24 - 8 * p;
    const unsigned int hmask = (p == 0) ? 0u : (0xFFFFFFFFu << (shift + 8));

    if (tid < 256u) hist[tid] = 0u;
    __syncthreads();

    #pragma unroll
    for (int j = 0; j < VPT; ++j) {
      const vf4 v = l4[tid + j * BLOCK];         // ds_load_b128
      const unsigned int k0 = f2key(v.x), k1 = f2key(v.y);
      const unsigned int k2 = f2key(v.z), k3 = f2key(v.w);
      if ((k0 & hmask) == prefix) atomicAdd(&hist[(k0 >> shift) & 0xFFu], 1u);
      if ((k1 & hmask) == prefix) atomicAdd(&hist[(k1 >> shift) & 0xFFu], 1u);
      if ((k2 & hmask) == prefix) atomicAdd(&hist[(k2 >> shift) & 0xFFu], 1u);
      if ((k3 & hmask) == prefix) atomicAdd(&hist[(k3 >> shift) & 0xFFu], 1u);
    }
    __syncthreads();

    if (tid < 256u) sfx[tid] = hist[tid];
    if (tid == 0u)  sfx[256] = 0u;
    __syncthreads();

    // Hillis-Steele suffix sums: sfx[b] = #keys in class with bin >= b
    for (int off = 1; off < 256; off <<= 1) {
      unsigned int v = 0u;
      if (tid < 256u) v = sfx[tid] + ((tid + off < 256u) ? sfx[tid + off] : 0u);
      __syncthreads();
      if (tid < 256u) sfx[tid] = v;
      __syncthreads();
    }

    if (tid < 256u) {
      const unsigned int hi = sfx[tid];
      const unsigned int lo = sfx[tid + 1];
      if (hi >= rem && lo < rem) {        // exactly one bin satisfies this
        bc[0] = prefix | (tid << shift);
        bc[1] = rem - lo;                 // rank within this bin
        bc[2] = hi - lo;                  // number of keys equal in this bin
      }
    }
    __syncthreads();
    prefix = bc[0];
    rem    = bc[1];
    __syncthreads();
  }

  const unsigned int T  = prefix;   // exact 32-bit key of the K-th largest element
  const unsigned int r  = rem;      // how many keys == T must be kept
  const unsigned int eq = bc[2];    // how many keys == T exist

  // ---------------- Stage 3: tie cutoff (lowest-index wins, matches lax.top_k) ----
  if (tid == 0u) {
    unsigned int cut = DIM - 1;
    if (r < eq) {                       // rare: duplicates straddle the boundary
      unsigned int c = 0;
      for (int i = 0; i < DIM; ++i) {
        if (f2key(row_f[i]) == T) {
          if (++c == r) { cut = (unsigned int)i; break; }
        }
      }
    }
    bc[3] = cut;
  }
  __syncthreads();
  const unsigned int cut = bc[3];

  // ---------------- Stage 4: masked write-out (b128, non-temporal) ----------------
  vf4* __restrict__ o4 = reinterpret_cast<vf4*>(out + (size_t)row * DIM);
  #pragma unroll
  for (int j = 0; j < VPT; ++j) {
    const int vi  = (int)tid + j * BLOCK;
    const unsigned int base = 4u * (unsigned int)vi;   // element index of .x
    const vf4 v = l4[vi];
    vf4 o;
    {
      const unsigned int k = f2key(v.x);
      o.x = ((k > T) || (k == T && (base + 0u) <= cut)) ? v.x : 0.0f;
    }
    {
      const unsigned int k = f2key(v.y);
      o.y = ((k > T) || (k == T && (base + 1u) <= cut)) ? v.y : 0.0f;
    }
    {
      const unsigned int k = f2key(v.z);
      o.z = ((k > T) || (k == T && (base + 2u) <= cut)) ? v.z : 0.0f;
    }
    {
      const unsigned int k = f2key(v.w);
      o.w = ((k > T) || (k == T && (base + 3u) <= cut)) ? v.w : 0.0f;
    }
    __builtin_nontemporal_store(o, &o4[vi]);           // global_store_b128, TH=NT
  }
}

extern "C" void kernel_launch(void* const* d_in, const int* in_sizes, int n_in,
                              void* d_out, int out_size, void* d_ws, size_t ws_size,
                              hipStream_t stream) {
  (void)n_in; (void)d_ws; (void)ws_size; (void)out_size;
  const float* x = (const float*)d_in[0];
  float* out = (float*)d_out;
  const int rows = in_sizes[0] / DIM;   // 4096
  topk_mask_kernel<<<dim3(rows), dim3(BLOCK), 0, stream>>>(x, out);
}